// DirectionalMambaBlock_34832184770991
// MI455X (gfx1250) — compile-verified
//
#include <hip/hip_runtime.h>
#include <math.h>

typedef __attribute__((ext_vector_type(2))) float v2f;
typedef __attribute__((ext_vector_type(8))) float v8f;

#define B_  4
#define L_  1024
#define DM  512      // D_MODEL
#define DI  1024     // D_INNER
#define NST 32       // D_STATE
#define DBC 96       // DT_RANK + 2*D_STATE

// snake ('tl_row' boustrophedon) source index for permuted position i
__device__ __forceinline__ int snake_src(int i) {
    int r = i >> 5;
    int c = i & 31;
    int cc = (r & 1) ? (31 - c) : c;
    return (r << 5) + cc;
}

// -------- 1. gather permutation: xp[b,i,:] = tokens[b, perm[i], :] --------
__global__ void permute_gather(const float* __restrict__ tokens,
                               float* __restrict__ xp, int total) {
    int idx = blockIdx.x * blockDim.x + threadIdx.x;
    if (idx >= total) return;
    int d    = idx & (DM - 1);
    int rest = idx >> 9;            // / DM
    int i    = rest & (L_ - 1);
    int b    = rest >> 10;
    int src  = snake_src(i);
    xp[idx] = tokens[((size_t)((b << 10) + src)) * DM + d];
}

// -------- generic f32 WMMA GEMM: C[m,n] = sum_k A[m,k]*W[n,k] (+epilogue) --------
// mode 0: plain store
// mode 1: softplus(acc + bias[n])                     (dt_proj -> delta)
// mode 2: gelu(acc + bias[n]) + resid[m,n], snake-scatter row  (final linear)
__global__ void gemm_wmma(const float* __restrict__ A, int lda,
                          const float* __restrict__ W, int ldw,
                          float* __restrict__ C, int ldc,
                          int M, int N, int K, int mode,
                          const float* __restrict__ bias,
                          const float* __restrict__ resid) {
    int wave = (blockIdx.x * blockDim.x + threadIdx.x) >> 5;
    int lane = threadIdx.x & 31;
    int tilesN = N >> 4;
    int tilesM = M >> 4;
    if (wave >= tilesM * tilesN) return;        // wave-uniform exit (EXEC all-1 for WMMA)
    int tm = wave / tilesN;
    int tn = wave - tm * tilesN;

    int mrow = (tm << 4) + (lane & 15);
    int ncol = (tn << 4) + (lane & 15);
    int kh   = lane >> 4;                       // lanes 0-15: K=0,1 ; lanes 16-31: K=2,3

    const float2* Ar = (const float2*)(A + (size_t)mrow * lda);
    const float2* Wr = (const float2*)(W + (size_t)ncol * ldw);

    v8f acc = {};
    int K2 = K >> 1;
    for (int k2 = 0; k2 < K2; k2 += 2) {        // K advances by 4 per WMMA
        float2 av = Ar[k2 + kh];
        float2 wv = Wr[k2 + kh];
        v2f a; a.x = av.x; a.y = av.y;
        v2f w; w.x = wv.x; w.y = wv.y;
        acc = __builtin_amdgcn_wmma_f32_16x16x4_f32(
            false, a, false, w, (short)0, acc, false, false);
    }

    int rbase = (lane >> 4) << 3;               // 0 or 8
    int col   = (tn << 4) + (lane & 15);
#pragma unroll
    for (int r = 0; r < 8; ++r) {
        int row = (tm << 4) + rbase + r;
        float v = acc[r];
        if (mode == 0) {
            C[(size_t)row * ldc + col] = v;
        } else if (mode == 1) {
            v += bias[col];
            float sp = fmaxf(v, 0.f) + log1pf(expf(-fabsf(v)));   // stable softplus
            C[(size_t)row * ldc + col] = sp;
        } else {
            v += bias[col];
            float g = 0.5f * v * (1.f + erff(v * 0.70710678118654752f)); // exact gelu
            g += resid[(size_t)row * DM + col];
            int b = row >> 10;
            int i = row & (L_ - 1);
            int dest = (b << 10) + snake_src(i);                  // out[:, inv, :]
            C[(size_t)dest * ldc + col] = g;
        }
    }
}

// -------- 3. causal depthwise conv (k=4) + SiLU --------
__global__ void conv_silu(const float* __restrict__ xz,      // [B,L,2048], xm = [...,0:1024]
                          const float* __restrict__ cw,      // [1024,4]
                          const float* __restrict__ cb,      // [1024]
                          float* __restrict__ u, int total) {
    int idx = blockIdx.x * blockDim.x + threadIdx.x;
    if (idx >= total) return;
    int d    = idx & (DI - 1);
    int rest = idx >> 10;
    int t    = rest & (L_ - 1);
    int b    = rest >> 10;
    float acc = cb[d];
#pragma unroll
    for (int j = 0; j < 4; ++j) {
        int tt = t - 3 + j;
        if (tt >= 0)
            acc += xz[((size_t)(b << 10) + tt) * 2048 + d] * cw[d * 4 + j];
    }
    u[idx] = acc / (1.f + expf(-acc));          // silu
}

// -------- 6. selective scan: one wave per (b,d); lane = state index n --------
__global__ void selective_scan(const float* __restrict__ delta,   // [B,L,DI]
                               const float* __restrict__ u,       // [B,L,DI]
                               const float* __restrict__ dbc,     // [B,L,96]
                               const float* __restrict__ xz,      // [B,L,2048], z = [...,1024:]
                               const float* __restrict__ A_log,   // [DI,32]
                               const float* __restrict__ Dp,      // [DI]
                               float* __restrict__ yg) {          // [B,L,DI]
    int wave = (blockIdx.x * blockDim.x + threadIdx.x) >> 5;
    int lane = threadIdx.x & 31;
    int b = wave >> 10;
    int d = wave & (DI - 1);
    float Adn = -expf(A_log[d * NST + lane]);
    float Dd  = Dp[d];
    float h = 0.f;
    size_t rowBase = (size_t)(b << 10);
    for (int t = 0; t < L_; ++t) {
        size_t row = rowBase + t;
        float dv = delta[row * DI + d];
        float uv = u[row * DI + d];
        float Bv = dbc[row * DBC + 32 + lane];
        float Cv = dbc[row * DBC + 64 + lane];
        float dA = expf(dv * Adn);
        h = dA * h + (dv * uv) * Bv;
        float y = h * Cv;
        for (int off = 16; off > 0; off >>= 1)
            y += __shfl_xor(y, off, 32);
        if (lane == 0) {
            float zv = xz[row * 2048 + 1024 + d];
            float sz = zv / (1.f + expf(-zv));
            yg[row * DI + d] = (y + uv * Dd) * sz;
        }
    }
}

// -------- 8. LayerNorm over rows of 512 --------
__global__ void layer_norm(const float* __restrict__ y2,
                           const float* __restrict__ lw,
                           const float* __restrict__ lb,
                           float* __restrict__ yn) {
    __shared__ float s1[256], s2[256];
    int row = blockIdx.x;
    int tid = threadIdx.x;
    float v0 = y2[(size_t)row * DM + tid];
    float v1 = y2[(size_t)row * DM + 256 + tid];
    s1[tid] = v0 + v1;
    s2[tid] = v0 * v0 + v1 * v1;
    __syncthreads();
    for (int s = 128; s > 0; s >>= 1) {
        if (tid < s) { s1[tid] += s1[tid + s]; s2[tid] += s2[tid + s]; }
        __syncthreads();
    }
    float mu  = s1[0] * (1.f / 512.f);
    float var = s2[0] * (1.f / 512.f) - mu * mu;
    float rs  = rsqrtf(var + 1e-5f);
    yn[(size_t)row * DM + tid]       = (v0 - mu) * rs * lw[tid]       + lb[tid];
    yn[(size_t)row * DM + 256 + tid] = (v1 - mu) * rs * lw[256 + tid] + lb[256 + tid];
}

static inline void launch_gemm(const float* A, int lda, const float* W, int ldw,
                               float* C, int ldc, int M, int N, int K, int mode,
                               const float* bias, const float* resid,
                               hipStream_t stream) {
    int tiles = (M >> 4) * (N >> 4);
    int grid  = (tiles + 7) / 8;                 // 8 waves / 256-thread block
    gemm_wmma<<<grid, 256, 0, stream>>>(A, lda, W, ldw, C, ldc, M, N, K, mode, bias, resid);
}

extern "C" void kernel_launch(void* const* d_in, const int* in_sizes, int n_in,
                              void* d_out, int out_size, void* d_ws, size_t ws_size,
                              hipStream_t stream) {
    (void)in_sizes; (void)n_in; (void)out_size; (void)ws_size;
    const float* tokens     = (const float*)d_in[0];
    const float* in_proj_w  = (const float*)d_in[1];
    const float* conv_w     = (const float*)d_in[2];
    const float* conv_b     = (const float*)d_in[3];
    const float* x_proj_w   = (const float*)d_in[4];
    const float* dt_proj_w  = (const float*)d_in[5];
    const float* dt_proj_b  = (const float*)d_in[6];
    const float* A_log      = (const float*)d_in[7];
    const float* Dp         = (const float*)d_in[8];
    const float* out_proj_w = (const float*)d_in[9];
    const float* ln_w       = (const float*)d_in[10];
    const float* ln_b       = (const float*)d_in[11];
    const float* lin_w      = (const float*)d_in[12];
    const float* lin_b      = (const float*)d_in[13];
    float* out = (float*)d_out;
    float* ws  = (float*)d_ws;

    const size_t ROWS = (size_t)B_ * L_;         // 4096
    float* xp    = ws;                            // [4096,512]
    float* xz    = xp    + ROWS * 2 * DM;         // [4096,2048]
    float* u     = xz    + ROWS * 4 * DM;         // [4096,1024]
    float* dbc   = u     + ROWS * DI;             // [4096,96]
    float* delta = dbc   + ROWS * DBC;            // [4096,1024]
    float* yg    = delta + ROWS * DI;             // [4096,1024]
    float* y2    = yg    + ROWS * DI;             // [4096,512]
    float* yn    = y2    + ROWS * DM;             // [4096,512]

    // 1. permute gather
    {
        int total = (int)(ROWS * DM);
        permute_gather<<<total / 256, 256, 0, stream>>>(tokens, xp, total);
    }
    // 2. in_proj: xz = xp @ in_proj_w.T    (M=4096, N=2048, K=512)
    launch_gemm(xp, DM, in_proj_w, DM, xz, 2 * DI, (int)ROWS, 2 * DI, DM, 0, nullptr, nullptr, stream);
    // 3. depthwise conv + silu -> u
    {
        int total = (int)(ROWS * DI);
        conv_silu<<<total / 256, 256, 0, stream>>>(xz, conv_w, conv_b, u, total);
    }
    // 4. x_proj: dbc = u @ x_proj_w.T      (M=4096, N=96, K=1024)
    launch_gemm(u, DI, x_proj_w, DI, dbc, DBC, (int)ROWS, DBC, DI, 0, nullptr, nullptr, stream);
    // 5. dt_proj + softplus: delta         (M=4096, N=1024, K=32; A = dt slice of dbc)
    launch_gemm(dbc, DBC, dt_proj_w, NST, delta, DI, (int)ROWS, DI, NST, 1, dt_proj_b, nullptr, stream);
    // 6. selective scan -> yg (fused +u*D and *silu(z))
    selective_scan<<<(B_ * DI * 32) / 256, 256, 0, stream>>>(delta, u, dbc, xz, A_log, Dp, yg);
    // 7. out_proj: y2 = yg @ out_proj_w.T  (M=4096, N=512, K=1024)
    launch_gemm(yg, DI, out_proj_w, DI, y2, DM, (int)ROWS, DM, DI, 0, nullptr, nullptr, stream);
    // 8. layernorm -> yn
    layer_norm<<<(int)ROWS, 256, 0, stream>>>(y2, ln_w, ln_b, yn);
    // 9. lin + gelu + residual + inverse-permute scatter -> out
    launch_gemm(yn, DM, lin_w, DM, out, DM, (int)ROWS, DM, DM, 2, lin_b, xp, stream);
}